// ConvCaps2D_77498389889571
// MI455X (gfx1250) — compile-verified
//
#include <hip/hip_runtime.h>
#include <math.h>

// ---- problem constants (from the reference) ----
#define BATCH   64
#define CIN     32
#define LIN     4096
#define OCH     256              // FILTERS(32) * CAPSLEN(8)
#define KS      5
#define KDIM    (CIN * KS)       // 160 = GEMM reduction dim
#define LOUT    2046             // (4096 - 5)/2 + 1
#define NT      128              // ceil(2046/16) N tiles
#define MBLK    4                // M-tiles per wave: 4 x 16 = 64 output channels
#define LDS_STRIDE 164           // padded A row stride: row*36 mod 64 -> rows own
                                 // disjoint 4-bank groups => conflict-free b64 reads

typedef __attribute__((ext_vector_type(2))) float v2f;
typedef __attribute__((ext_vector_type(8))) float v8f;

// K permutation (A in LDS and implicit im2col B use the same one):
//   taps 0..3: K-step = [c,2p],[c,2p+1],[c+1,2p],[c+1,2p+1]  (tap-pair p, channel-pair c)
//              -> each lane's B fragment comes straight out of a b128 load of taps 0..3
//                 (pairs land register-adjacent, zero movs); half-dependence is a
//                 constant +LIN folded into the base pointer.
//   tap 4    : tap-major tail, channels c0..c0+3 (b32 pair at stride LIN).
// k(c,kk) = kk<4 ? (kk>>1)*64 + 2*c + (kk&1) : 128 + c

__global__ __launch_bounds__(256) void convcaps_wmma_f32(
    const float* __restrict__ x,     // [64, 32, 4096]
    const float* __restrict__ W,     // [256, 32, 5]
    const float* __restrict__ bias,  // [256]
    float* __restrict__ out)         // [64, 32*2046, 8]
{
    __shared__ float As[16 * MBLK * LDS_STRIDE];   // 64 rows x 164 floats = 42 KB

    const int mg    = blockIdx.y;        // 0..3  -> 64-channel M group
    const int batch = blockIdx.z;        // 0..63
    const int m0    = mg * (16 * MBLK);

    // ---- cooperative stage of A tile (64 x 160) into LDS in permuted K order ----
    for (int idx = threadIdx.x; idx < 16 * MBLK * KDIM; idx += 256) {
        const int row = idx / KDIM;
        const int t   = idx - row * KDIM;     // t = c*5 + kk (memory order of W)
        const int c   = t / KS;
        const int kk  = t - c * KS;
        const int col = (kk < 4) ? ((kk >> 1) * 64 + 2 * c + (kk & 1)) : (128 + c);
        As[row * LDS_STRIDE + col] = W[(m0 + row) * KDIM + t];
    }
    __syncthreads();

    const int wave = threadIdx.x >> 5;
    const int lane = threadIdx.x & 31;
    const int half = lane >> 4;          // 0: lanes 0-15, 1: lanes 16-31
    const int l15  = lane & 15;

    const int nt = blockIdx.x * 8 + wave;            // 0..127
    const int n  = nt * 16 + l15;                    // output position this lane owns
    const int nc = (n < LOUT - 1) ? n : (LOUT - 1);  // clamp loads in-bounds

    const float* __restrict__ xp0 = x + (size_t)batch * (CIN * LIN) + 2 * nc;
    const float* __restrict__ xq  = xp0 + half * LIN;        // clean part: +1 channel
    const float* __restrict__ xt  = xp0 + half * (2 * LIN);  // tail part:  +2 channels

    // A fragment sources, one per M-tile: lanes 0-15 read K={k0,k0+1}, 16-31 K={k0+2,k0+3}
    const float* asrc[MBLK];
#pragma unroll
    for (int t = 0; t < MBLK; ++t)
        asrc[t] = &As[(t * 16 + l15) * LDS_STRIDE + 2 * half];

    v8f acc[MBLK] = {};

    // ---- taps 0..3: 32 K-steps, B fragments straight out of b128-merged loads ----
#pragma unroll
    for (int i = 0; i < 32; ++i) {
        const int k0 = 4 * i;
        const int p  = k0 >> 6;            // tap-pair 0 or 1   (compile-time)
        const int cb = (k0 & 63) >> 1;     // channel-pair base (compile-time)

        v2f bb = *(const v2f*)(xq + cb * LIN + 2 * p);   // taps {2p,2p+1} of my channel

#pragma unroll
        for (int t = 0; t < MBLK; ++t) {
            v2f a = *(const v2f*)(asrc[t] + k0);         // conflict-free ds b64
            acc[t] = __builtin_amdgcn_wmma_f32_16x16x4_f32(
                false, a, false, bb, (short)0, acc[t], false, false);
        }
    }

    // ---- tap 4 tail: 8 K-steps, tap-major (channels c0..c0+3) ----
#pragma unroll
    for (int i = 0; i < 8; ++i) {
        const int k0 = 128 + 4 * i;
        const int c0 = 4 * i;

        v2f bb;
        bb.x = xt[c0 * LIN + 4];
        bb.y = xt[c0 * LIN + 4 + LIN];

#pragma unroll
        for (int t = 0; t < MBLK; ++t) {
            v2f a = *(const v2f*)(asrc[t] + k0);
            acc[t] = __builtin_amdgcn_wmma_f32_16x16x4_f32(
                false, a, false, bb, (short)0, acc[t], false, false);
        }
    }

    // C layout: in tile t, VGPR j holds channel m0 + t*16 + 8*half + j at position n.
    // => each accumulator is one full capsule vector: bias + squash lane-locally.
#pragma unroll
    for (int t = 0; t < MBLK; ++t) {
        const int f = (m0 + t * 16) / 8 + half;            // capsule index 0..31
        const float4* bptr = (const float4*)(bias + m0 + t * 16 + 8 * half);
        const float4 b0 = bptr[0];
        const float4 b1 = bptr[1];

        float v0 = acc[t][0] + b0.x, v1 = acc[t][1] + b0.y;
        float v2 = acc[t][2] + b0.z, v3 = acc[t][3] + b0.w;
        float v4 = acc[t][4] + b1.x, v5 = acc[t][5] + b1.y;
        float v6 = acc[t][6] + b1.z, v7 = acc[t][7] + b1.w;

        const float norm = v0*v0 + v1*v1 + v2*v2 + v3*v3 +
                           v4*v4 + v5*v5 + v6*v6 + v7*v7;
        // squash factor: (norm/(1+norm))/sqrt(norm) == sqrt(norm)*rcp(1+norm)
        // hardware TRANS ops: v_sqrt_f32 + v_rcp_f32 (2 instrs vs ~30 IEEE-expanded)
        const float s = __builtin_amdgcn_sqrtf(norm) *
                        __builtin_amdgcn_rcpf(1.0f + norm);

        if (n < LOUT) {
            const size_t o = ((size_t)(batch * 32 + f) * LOUT + n) * 8;
            float4 r0, r1;
            r0.x = v0 * s; r0.y = v1 * s; r0.z = v2 * s; r0.w = v3 * s;
            r1.x = v4 * s; r1.y = v5 * s; r1.z = v6 * s; r1.w = v7 * s;
            *(float4*)(out + o)     = r0;   // 32B/lane, coalesced b128 stores
            *(float4*)(out + o + 4) = r1;
        }
    }
}

extern "C" void kernel_launch(void* const* d_in, const int* in_sizes, int n_in,
                              void* d_out, int out_size, void* d_ws, size_t ws_size,
                              hipStream_t stream) {
    const float* x    = (const float*)d_in[0];
    const float* W    = (const float*)d_in[1];
    const float* bias = (const float*)d_in[2];
    float* out        = (float*)d_out;

    dim3 block(256);                          // 8 waves, each: 4 M-tiles x 1 N-tile
    dim3 grid(NT / 8, OCH / (16 * MBLK), BATCH);
    hipLaunchKernelGGL(convcaps_wmma_f32, grid, block, 0, stream, x, W, bias, out);
}